// HierarchicalGNNBlock_43568148250791
// MI455X (gfx1250) — compile-verified
//
#include <hip/hip_runtime.h>
#include <cmath>
#include <cstdint>

typedef float v2f __attribute__((ext_vector_type(2)));
typedef float v8f __attribute__((ext_vector_type(8)));
typedef unsigned v4u __attribute__((ext_vector_type(4)));
typedef int v4i __attribute__((ext_vector_type(4)));
typedef int v8i __attribute__((ext_vector_type(8)));

#define Nn 50000
#define Ee 200000
#define Ll 128
#define Ss 512
#define CH 32  // K-chunk rows staged per TDM transfer

static __device__ __forceinline__ v8f wmma_f32(v2f a, v2f b, v8f c) {
  // V_WMMA_F32_16X16X4_F32: D = A(16x4) x B(4x16) + C, exact fp32
  return __builtin_amdgcn_wmma_f32_16x16x4_f32(false, a, false, b, (short)0, c,
                                               false, false);
}

// ---------------------------------------------------------------------------
// Tensor Data Mover: stage a rows x cols fp32 tile of W (row-major, leading
// dim ld elements) starting at (r0, c0) into LDS at ldsDst.  Rows past
// totRows read as zero (TDM OOB semantics) -> free zero padding of ragged K.
// D# packing per CDNA5 ISA.
// ---------------------------------------------------------------------------
static __device__ __forceinline__ void tdm_stage_f32(const float* W, int ld,
                                                     int totRows, int r0,
                                                     int c0, int rows,
                                                     int cols, float* ldsDst) {
  unsigned lds_addr = (unsigned)(uintptr_t)ldsDst;  // aperture low 32 = offset
  unsigned long long g =
      (unsigned long long)(uintptr_t)(W + (size_t)r0 * ld + c0);
  unsigned long long td0 = (unsigned)cols;           // tensor_dim0
  unsigned long long td1 = (unsigned)(totRows - r0); // remaining rows
  unsigned long long st0 = (unsigned)ld;             // row stride (elements)
  v4u g0;
  g0.x = 1u;                                          // count=1 (user D#)
  g0.y = lds_addr;                                    // lds_addr [63:32]
  g0.z = (unsigned)g;                                 // global_addr lo
  g0.w = ((unsigned)(g >> 32) & 0x01FFFFFFu) | 0x80000000u;  // addr hi|type=2
  v8i g1;
  g1[0] = 0x00020000;                                 // data_size=4B
  g1[1] = (int)((td0 & 0xFFFFull) << 16);             // tensor_dim0[15:0]
  g1[2] = (int)(((td0 >> 16) & 0xFFFFull) | ((td1 & 0xFFFFull) << 16));
  g1[3] = (int)(((td1 >> 16) & 0xFFFFull) | ((unsigned)cols << 16)); // tile0
  g1[4] = rows & 0xFFFF;                              // tile_dim1
  g1[5] = (int)(st0 & 0xFFFFFFFFull);                 // stride0[31:0]
  g1[6] = (int)((st0 >> 32) & 0xFFFFull);             // stride0[47:32]
  g1[7] = 0;
  v4i z4 = {0, 0, 0, 0};
  v8i z8 = {0, 0, 0, 0, 0, 0, 0, 0};
  __builtin_amdgcn_tensor_load_to_lds(g0, g1, z4, z4, z8, 0);
}

// Load a contiguous fp32 pair (c, c+1) of virtual concat of 3 width-128
// segments (optional row gather).  Pairs never straddle segments (c even).
static __device__ __forceinline__ v2f xload2(int row, int c, const float* p0,
                                             const int* i0, const float* p1,
                                             const int* i1, const float* p2,
                                             const int* i2) {
  const float* p;
  const int* idx;
  if (c < 128) {
    p = p0;
    idx = i0;
  } else if (c < 256) {
    p = p1;
    idx = i1;
    c -= 128;
  } else {
    p = p2;
    idx = i2;
    c -= 256;
  }
  int r = idx ? idx[row] : row;
  return *(const v2f*)(p + (size_t)r * 128 + c);
}

// ---------------------------------------------------------------------------
// Stage-1 GEMM: C[M,128] = A[M,K] @ B[K,128]; B TDM-staged in CH-row chunks.
// 2 waves/block, 32 rows/wave (two A tiles sharing every B operand).
// ---------------------------------------------------------------------------
__global__ __launch_bounds__(64) void gemm_nn_128(int M, int K,
                                                  const float* __restrict__ A,
                                                  const float* __restrict__ B,
                                                  float* __restrict__ C) {
  __shared__ float wchunk[CH * 128];
  const int lane = threadIdx.x & 31;
  const int wave = threadIdx.x >> 5;
  const int r0 = (blockIdx.x * 2 + wave) * 32;
  const int nl = lane & 15;
  const int hi = lane >> 4;
  const int koff = hi * 2;
  const int madd = hi * 8;
  const int rc0 = (r0 + nl < M) ? r0 + nl : M - 1;
  const int rc1 = (r0 + 16 + nl < M) ? r0 + 16 + nl : M - 1;

  v8f acc[2][8] = {};
  for (int kc = 0; kc < K; kc += CH) {
    __syncthreads();
    if (wave == 0) {
      tdm_stage_f32(B, 128, K, kc, 0, CH, 128, wchunk);
      __builtin_amdgcn_s_wait_tensorcnt(0);
    }
    __syncthreads();
    for (int kk = 0; kk < CH; kk += 4) {
      int c0 = kc + kk + koff;
      v2f a0 = *(const v2f*)(A + (size_t)rc0 * K + c0);
      v2f a1 = *(const v2f*)(A + (size_t)rc1 * K + c0);
      for (int t = 0; t < 8; t++) {
        v2f b;
        int n = t * 16 + nl;
        b.x = wchunk[(kk + koff) * 128 + n];
        b.y = wchunk[(kk + koff + 1) * 128 + n];
        acc[0][t] = wmma_f32(a0, b, acc[0][t]);
        acc[1][t] = wmma_f32(a1, b, acc[1][t]);
      }
    }
  }
  for (int st = 0; st < 2; st++)
    for (int t = 0; t < 8; t++) {
      int n = t * 16 + nl;
      for (int v = 0; v < 8; v++) {
        int row = r0 + st * 16 + v + madd;
        if (row < M) C[(size_t)row * 128 + n] = acc[st][t][v];
      }
    }
}

// ---------------------------------------------------------------------------
// Fused 2-layer MLP: out = act2(LN2(relu(LN1(X@W1+b1))@W2+b2)) + resid
// X = concat of three width-128 segments (optional row gather).  Weights
// TDM-staged; layer-1 activations re-staged via bank-padded LDS for layer 2.
// 2 waves/block, 32 rows/wave.
// ---------------------------------------------------------------------------
__global__ __launch_bounds__(64) void mlp2_kernel(
    int M, int din, const float* p0, const int* i0, const float* p1,
    const int* i1, const float* p2, const int* i2,
    const float* __restrict__ W1, const float* __restrict__ b1,
    const float* __restrict__ g1, const float* __restrict__ be1,
    const float* __restrict__ W2, const float* __restrict__ b2,
    const float* __restrict__ g2, const float* __restrict__ be2,
    const float* __restrict__ resid, float* __restrict__ out, int act_tanh) {
  __shared__ float wchunk[CH * 128];
  __shared__ float hbuf_s[2 * 32 * 132];
  const int lane = threadIdx.x & 31;
  const int wave = threadIdx.x >> 5;
  const int r0 = (blockIdx.x * 2 + wave) * 32;
  const int nl = lane & 15;
  const int hi = lane >> 4;
  const int koff = hi * 2;
  const int madd = hi * 8;
  const int rc0 = (r0 + nl < M) ? r0 + nl : M - 1;
  const int rc1 = (r0 + 16 + nl < M) ? r0 + 16 + nl : M - 1;
  float* hbuf = hbuf_s + wave * (32 * 132);

  // ---- layer 1: X @ W1 ----
  v8f acc[2][8] = {};
  for (int kc = 0; kc < din; kc += CH) {
    __syncthreads();
    if (wave == 0) {
      tdm_stage_f32(W1, 128, din, kc, 0, CH, 128, wchunk);
      __builtin_amdgcn_s_wait_tensorcnt(0);
    }
    __syncthreads();
    int kmax = din - kc;
    if (kmax > CH) kmax = CH;
    for (int kk = 0; kk < kmax; kk += 4) {
      int c0 = kc + kk + koff;
      v2f a0 = xload2(rc0, c0, p0, i0, p1, i1, p2, i2);
      v2f a1 = xload2(rc1, c0, p0, i0, p1, i1, p2, i2);
      for (int t = 0; t < 8; t++) {
        v2f b;
        int n = t * 16 + nl;
        b.x = wchunk[(kk + koff) * 128 + n];
        b.y = wchunk[(kk + koff + 1) * 128 + n];
        acc[0][t] = wmma_f32(a0, b, acc[0][t]);
        acc[1][t] = wmma_f32(a1, b, acc[1][t]);
      }
    }
  }
  // ---- bias + LN + relu -> hbuf (A-layout restage) ----
  for (int st = 0; st < 2; st++) {
    float gl[8], bl[8];
    for (int t = 0; t < 8; t++) {
      int n = t * 16 + nl;
      gl[t] = g1[n];
      bl[t] = be1[n];
      acc[st][t] += b1[n];
    }
    for (int v = 0; v < 8; v++) {
      float s = 0.f, sq = 0.f;
      for (int t = 0; t < 8; t++) {
        float xx = acc[st][t][v];
        s += xx;
        sq += xx * xx;
      }
      for (int m = 1; m < 16; m <<= 1) {
        s += __shfl_xor(s, m, 32);
        sq += __shfl_xor(sq, m, 32);
      }
      float mean = s * (1.f / 128.f);
      float var = sq * (1.f / 128.f) - mean * mean;
      float rstd = rsqrtf(var + 1e-5f);
      for (int t = 0; t < 8; t++) {
        float xx = (acc[st][t][v] - mean) * rstd * gl[t] + bl[t];
        hbuf[(st * 16 + v + madd) * 132 + t * 16 + nl] = fmaxf(xx, 0.f);
      }
    }
  }
  // ---- layer 2: h @ W2 (A from LDS, ds_load_b64) ----
  v8f acc2[2][8] = {};
  for (int kc = 0; kc < 128; kc += CH) {
    __syncthreads();
    if (wave == 0) {
      tdm_stage_f32(W2, 128, 128, kc, 0, CH, 128, wchunk);
      __builtin_amdgcn_s_wait_tensorcnt(0);
    }
    __syncthreads();
    for (int kk = 0; kk < CH; kk += 4) {
      v2f a0 = *(const v2f*)&hbuf[nl * 132 + kc + kk + koff];
      v2f a1 = *(const v2f*)&hbuf[(16 + nl) * 132 + kc + kk + koff];
      for (int t = 0; t < 8; t++) {
        v2f b;
        int n = t * 16 + nl;
        b.x = wchunk[(kk + koff) * 128 + n];
        b.y = wchunk[(kk + koff + 1) * 128 + n];
        acc2[0][t] = wmma_f32(a0, b, acc2[0][t]);
        acc2[1][t] = wmma_f32(a1, b, acc2[1][t]);
      }
    }
  }
  // ---- bias + LN + act + residual -> out ----
  for (int st = 0; st < 2; st++) {
    float gl[8], bl[8];
    for (int t = 0; t < 8; t++) {
      int n = t * 16 + nl;
      gl[t] = g2[n];
      bl[t] = be2[n];
      acc2[st][t] += b2[n];
    }
    for (int v = 0; v < 8; v++) {
      float s = 0.f, sq = 0.f;
      for (int t = 0; t < 8; t++) {
        float xx = acc2[st][t][v];
        s += xx;
        sq += xx * xx;
      }
      for (int m = 1; m < 16; m <<= 1) {
        s += __shfl_xor(s, m, 32);
        sq += __shfl_xor(sq, m, 32);
      }
      float mean = s * (1.f / 128.f);
      float var = sq * (1.f / 128.f) - mean * mean;
      float rstd = rsqrtf(var + 1e-5f);
      int row = r0 + st * 16 + v + madd;
      if (row < M) {
        for (int t = 0; t < 8; t++) {
          float xx = (acc2[st][t][v] - mean) * rstd * gl[t] + bl[t];
          xx = act_tanh ? tanhf(xx) : fmaxf(xx, 0.f);
          int n = t * 16 + nl;
          out[(size_t)row * 128 + n] = xx + resid[(size_t)row * 128 + n];
        }
      }
    }
  }
}

// ---------------------------------------------------------------------------
// Assignment net: h = relu(LN(cat(nodes,x)[N,131] @ W1 + b1));
// unnorm = h @ W2[128,512] + b2.  Ragged K=131 zero-padded by TDM OOB reads.
// ---------------------------------------------------------------------------
__global__ __launch_bounds__(64) void assign_kernel(
    int M, const float* __restrict__ nodes, const float* __restrict__ xin,
    const float* __restrict__ W1, const float* __restrict__ b1,
    const float* __restrict__ g1, const float* __restrict__ be1,
    const float* __restrict__ W2, const float* __restrict__ b2,
    float* __restrict__ unnorm) {
  __shared__ float wchunk[CH * 128];
  __shared__ float hbuf_s[2 * 32 * 132];
  const int lane = threadIdx.x & 31;
  const int wave = threadIdx.x >> 5;
  const int r0 = (blockIdx.x * 2 + wave) * 32;
  const int nl = lane & 15;
  const int hi = lane >> 4;
  const int koff = hi * 2;
  const int madd = hi * 8;
  const int rc0 = (r0 + nl < M) ? r0 + nl : M - 1;
  const int rc1 = (r0 + 16 + nl < M) ? r0 + 16 + nl : M - 1;
  float* hbuf = hbuf_s + wave * (32 * 132);
  const int din = 131;  // L + XD

  v8f acc[2][8] = {};
  for (int kc = 0; kc < din; kc += CH) {
    __syncthreads();
    if (wave == 0) {
      tdm_stage_f32(W1, 128, din, kc, 0, CH, 128, wchunk);  // OOB rows -> 0
      __builtin_amdgcn_s_wait_tensorcnt(0);
    }
    __syncthreads();
    int kmax = din - kc;
    if (kmax > CH) kmax = CH;
    for (int kk = 0; kk < kmax; kk += 4) {
      int c0 = kc + kk + koff, c1 = c0 + 1;
      v2f a0, a1;
      a0.x = (c0 < 128) ? nodes[(size_t)rc0 * 128 + c0]
                        : ((c0 < din) ? xin[(size_t)rc0 * 3 + c0 - 128] : 0.f);
      a0.y = (c1 < 128) ? nodes[(size_t)rc0 * 128 + c1]
                        : ((c1 < din) ? xin[(size_t)rc0 * 3 + c1 - 128] : 0.f);
      a1.x = (c0 < 128) ? nodes[(size_t)rc1 * 128 + c0]
                        : ((c0 < din) ? xin[(size_t)rc1 * 3 + c0 - 128] : 0.f);
      a1.y = (c1 < 128) ? nodes[(size_t)rc1 * 128 + c1]
                        : ((c1 < din) ? xin[(size_t)rc1 * 3 + c1 - 128] : 0.f);
      for (int t = 0; t < 8; t++) {
        v2f b;  // rows >= kmax are zero-filled by TDM, no guard needed
        int n = t * 16 + nl;
        b.x = wchunk[(kk + koff) * 128 + n];
        b.y = wchunk[(kk + koff + 1) * 128 + n];
        acc[0][t] = wmma_f32(a0, b, acc[0][t]);
        acc[1][t] = wmma_f32(a1, b, acc[1][t]);
      }
    }
  }
  for (int st = 0; st < 2; st++) {
    float gl[8], bl[8];
    for (int t = 0; t < 8; t++) {
      int n = t * 16 + nl;
      gl[t] = g1[n];
      bl[t] = be1[n];
      acc[st][t] += b1[n];
    }
    for (int v = 0; v < 8; v++) {
      float s = 0.f, sq = 0.f;
      for (int t = 0; t < 8; t++) {
        float xx = acc[st][t][v];
        s += xx;
        sq += xx * xx;
      }
      for (int m = 1; m < 16; m <<= 1) {
        s += __shfl_xor(s, m, 32);
        sq += __shfl_xor(sq, m, 32);
      }
      float mean = s * (1.f / 128.f);
      float var = sq * (1.f / 128.f) - mean * mean;
      float rstd = rsqrtf(var + 1e-5f);
      for (int t = 0; t < 8; t++) {
        float xx = (acc[st][t][v] - mean) * rstd * gl[t] + bl[t];
        hbuf[(st * 16 + v + madd) * 132 + t * 16 + nl] = fmaxf(xx, 0.f);
      }
    }
  }
  // layer 2: dout=512 in 4 column blocks of 128; W2 block TDM-staged
  for (int cc = 0; cc < 4; cc++) {
    v8f acc2[2][8] = {};
    for (int kc = 0; kc < 128; kc += CH) {
      __syncthreads();
      if (wave == 0) {
        tdm_stage_f32(W2, 512, 128, kc, cc * 128, CH, 128, wchunk);
        __builtin_amdgcn_s_wait_tensorcnt(0);
      }
      __syncthreads();
      for (int kk = 0; kk < CH; kk += 4) {
        v2f a0 = *(const v2f*)&hbuf[nl * 132 + kc + kk + koff];
        v2f a1 = *(const v2f*)&hbuf[(16 + nl) * 132 + kc + kk + koff];
        for (int t = 0; t < 8; t++) {
          v2f b;
          int n = t * 16 + nl;
          b.x = wchunk[(kk + koff) * 128 + n];
          b.y = wchunk[(kk + koff + 1) * 128 + n];
          acc2[0][t] = wmma_f32(a0, b, acc2[0][t]);
          acc2[1][t] = wmma_f32(a1, b, acc2[1][t]);
        }
      }
    }
    for (int st = 0; st < 2; st++)
      for (int t = 0; t < 8; t++) {
        int n = cc * 128 + t * 16 + nl;
        float bv = b2[n];
        for (int v = 0; v < 8; v++) {
          int row = r0 + st * 16 + v + madd;
          if (row < M) unnorm[(size_t)row * 512 + n] = acc2[st][t][v] + bv;
        }
      }
  }
}

// ---------------------------------------------------------------------------
// Row softmax over 512 columns; one wave per row (16 values/lane).
// ---------------------------------------------------------------------------
__global__ __launch_bounds__(128) void softmax512(const float* __restrict__ in,
                                                  float* __restrict__ out,
                                                  int M) {
  const int lane = threadIdx.x & 31;
  const int wave = threadIdx.x >> 5;
  const int row = blockIdx.x * 4 + wave;
  if (row >= M) return;
  const float* p = in + (size_t)row * 512;
  float v[16];
  float mx = -3.402823466e+38f;
  for (int i = 0; i < 16; i++) {
    v[i] = p[lane + 32 * i];
    mx = fmaxf(mx, v[i]);
  }
  for (int m = 1; m < 32; m <<= 1) mx = fmaxf(mx, __shfl_xor(mx, m, 32));
  float s = 0.f;
  for (int i = 0; i < 16; i++) {
    v[i] = __expf(v[i] - mx);
    s += v[i];
  }
  for (int m = 1; m < 32; m <<= 1) s += __shfl_xor(s, m, 32);
  float inv = 1.f / s;
  float* o = out + (size_t)row * 512;
  for (int i = 0; i < 16; i++) o[lane + 32 * i] = v[i] * inv;
}

// ---------------------------------------------------------------------------
// node_msg[512,128] += assignment[N,512]^T @ nodes[N,128], split-K over N.
// 16 waves/block; wave w owns s-rows [32w, 32w+32); chunk of 512 nodes/block.
// ---------------------------------------------------------------------------
__global__ __launch_bounds__(512) void at_gemm(const float* __restrict__ assign,
                                               const float* __restrict__ nodes,
                                               float* __restrict__ node_msg,
                                               int N) {
  const int lane = threadIdx.x & 31;
  const int wave = threadIdx.x >> 5;
  const int nl = lane & 15;
  const int hi = lane >> 4;
  const int koff = hi * 2;
  const int madd = hi * 8;
  const int kbeg = blockIdx.x * 512;
  const int kend = (kbeg + 512 < N) ? (kbeg + 512) : N;
  const int s0 = wave * 32;

  v8f acc[2][8] = {};
  for (int k = kbeg; k < kend; k += 4) {
    int kk0 = k + koff, kk1 = kk0 + 1;
    v2f a0, a1;
    a0.x = (kk0 < kend) ? assign[(size_t)kk0 * 512 + s0 + nl] : 0.f;
    a0.y = (kk1 < kend) ? assign[(size_t)kk1 * 512 + s0 + nl] : 0.f;
    a1.x = (kk0 < kend) ? assign[(size_t)kk0 * 512 + s0 + 16 + nl] : 0.f;
    a1.y = (kk1 < kend) ? assign[(size_t)kk1 * 512 + s0 + 16 + nl] : 0.f;
    for (int t = 0; t < 8; t++) {
      v2f b;
      int n = t * 16 + nl;
      b.x = (kk0 < kend) ? nodes[(size_t)kk0 * 128 + n] : 0.f;
      b.y = (kk1 < kend) ? nodes[(size_t)kk1 * 128 + n] : 0.f;
      acc[0][t] = wmma_f32(a0, b, acc[0][t]);
      acc[1][t] = wmma_f32(a1, b, acc[1][t]);
    }
  }
  for (int st = 0; st < 2; st++)
    for (int t = 0; t < 8; t++) {
      int n = t * 16 + nl;
      for (int v = 0; v < 8; v++) {
        int s = s0 + st * 16 + v + madd;
        atomicAdd(&node_msg[(size_t)s * 128 + n], acc[st][t][v]);
      }
    }
}

// ---------------------------------------------------------------------------
// segment_sum: msg[dst[e], :] += edges[e, :]; one wave per edge, float4/lane.
// ---------------------------------------------------------------------------
__global__ __launch_bounds__(256) void segsum_kernel(
    const float* __restrict__ edges, const int* __restrict__ dst,
    float* __restrict__ msg, int E) {
  int tid = blockIdx.x * blockDim.x + threadIdx.x;
  int e = tid >> 5;
  if (e >= E) return;
  int c = (tid & 31) * 4;
  int d = dst[e];
  const float* src = edges + (size_t)e * 128 + c;
  float* base = msg + (size_t)d * 128 + c;
  atomicAdd(base + 0, src[0]);
  atomicAdd(base + 1, src[1]);
  atomicAdd(base + 2, src[2]);
  atomicAdd(base + 3, src[3]);
}

extern "C" void kernel_launch(void* const* d_in, const int* in_sizes, int n_in,
                              void* d_out, int out_size, void* d_ws,
                              size_t ws_size, hipStream_t stream) {
  const float* x = (const float*)d_in[0];
  const float* nodes = (const float*)d_in[1];
  const float* edges = (const float*)d_in[2];
  const float* assignment = (const float*)d_in[3];
  const float* supernodes = (const float*)d_in[4];
  const int* graph = (const int*)d_in[5];
  const int* gsrc = graph;
  const int* gdst = graph + Ee;

  // params flattened leaf-order: node1[8], node2[8], super[8], edge[8],
  // assign[6] = aW1, ab1, ag, abe, aW2, ab2
  const float* prm[38];
  for (int i = 0; i < 38; i++) prm[i] = (const float*)d_in[6 + i];
  const float** n1 = prm + 0;
  const float** n2 = prm + 8;
  const float** sp = prm + 16;
  const float** ed = prm + 24;
  const float** as = prm + 32;

  float* out = (float*)d_out;
  float* out_nodes2 = out;                           // N*L
  float* out_edges2 = out_nodes2 + (size_t)Nn * Ll;  // E*L
  float* out_assign = out_edges2 + (size_t)Ee * Ll;  // N*S
  float* out_super = out_assign + (size_t)Nn * Ss;   // S*L
  float* out_unnorm = out_super + (size_t)Ss * Ll;   // N*S

  float* tmpNL = (float*)d_ws;                 // N*128, reused twice
  float* node_msg = tmpNL + (size_t)Nn * 128;  // S*128
  float* nodes1 = out_nodes2;  // nodes1 lives in-place in the nodes output

  const int nblkN = (Nn + 63) / 64;  // 782
  const int nblkE = (Ee + 63) / 64;  // 3125
  const int nblkS = (Ss + 63) / 64;  // 8

  // 1) sn_msg = assignment @ supernodes              (tmpNL)
  gemm_nn_128<<<nblkN, 64, 0, stream>>>(Nn, Ss, assignment, supernodes, tmpNL);

  // 2) nodes1 = MLP(node1)(cat(nodes, sn_msg)) + nodes
  mlp2_kernel<<<nblkN, 64, 0, stream>>>(
      Nn, 256, nodes, nullptr, tmpNL, nullptr, nodes, nullptr, n1[0], n1[1],
      n1[2], n1[3], n1[4], n1[5], n1[6], n1[7], nodes, nodes1, 0);

  // 3) unnorm = assign_net(cat(nodes1, x)); assignment2 = softmax(unnorm)
  assign_kernel<<<nblkN, 64, 0, stream>>>(Nn, nodes1, x, as[0], as[1], as[2],
                                          as[3], as[4], as[5], out_unnorm);
  softmax512<<<(Nn + 3) / 4, 128, 0, stream>>>(out_unnorm, out_assign, Nn);

  // 4) node_msg = assignment2^T @ nodes1 (split-K + atomics)
  (void)hipMemsetAsync(node_msg, 0, (size_t)Ss * 128 * sizeof(float), stream);
  at_gemm<<<(Nn + 511) / 512, 512, 0, stream>>>(out_assign, nodes1, node_msg,
                                                Nn);

  // 5) supernodes2 = MLP(super)(cat(supernodes, node_msg)) + supernodes
  mlp2_kernel<<<nblkS, 64, 0, stream>>>(
      Ss, 256, supernodes, nullptr, node_msg, nullptr, supernodes, nullptr,
      sp[0], sp[1], sp[2], sp[3], sp[4], sp[5], sp[6], sp[7], supernodes,
      out_super, 0);

  // 6) edges2 = MLP(edge)(cat(nodes1[src], nodes1[dst], edges)) tanh + edges
  mlp2_kernel<<<nblkE, 64, 0, stream>>>(
      Ee, 384, nodes1, gsrc, nodes1, gdst, edges, nullptr, ed[0], ed[1], ed[2],
      ed[3], ed[4], ed[5], ed[6], ed[7], edges, out_edges2, 1);

  // 7) edge_msg = segment_sum(edges2, dst)           (tmpNL reused)
  (void)hipMemsetAsync(tmpNL, 0, (size_t)Nn * 128 * sizeof(float), stream);
  segsum_kernel<<<(Ee * 32 + 255) / 256, 256, 0, stream>>>(out_edges2, gdst,
                                                           tmpNL, Ee);

  // 8) nodes2 = MLP(node2)(cat(nodes1, edge_msg)) + nodes1  (in-place safe:
  //    every read (segments + residual) is row-local to the owning wave)
  mlp2_kernel<<<nblkN, 64, 0, stream>>>(
      Nn, 256, nodes1, nullptr, tmpNL, nullptr, nodes1, nullptr, n2[0], n2[1],
      n2[2], n2[3], n2[4], n2[5], n2[6], n2[7], nodes1, out_nodes2, 0);
}